// sim_57904749084724
// MI455X (gfx1250) — compile-verified
//
#include <hip/hip_runtime.h>

#define N_NODES 100000
#define IN_CH   128      // HEADS*OUT_CH == IN_CH, heads are layout-transparent
#define N_EDGES 1600000

typedef float v2f __attribute__((ext_vector_type(2)));
typedef float v8f __attribute__((ext_vector_type(8)));

// Guarantee the hardware global_atomic_add_f32 path (no CAS fallback).
__device__ __forceinline__ void fatomic_add(float* p, float v) {
    unsafeAtomicAdd(p, v);
}

// ---------------------------------------------------------------------------
// y[m][n] = sum_k X[m][k] * W[n][k]  (+ bias[n] if bias != nullptr)
// Shapes: X [nrows x 128], W [128 x 128], Y [nrows x 128].
// Block = 256 threads = 8 waves; wave w owns output columns [16w, 16w+16).
// Each wave preloads its whole-K B strip into 64 VGPRs (reused across all row
// tiles via grid-stride loop), A tiles staged in padded LDS.
// Uses V_WMMA_F32_16X16X4_F32: A is 16x4, B is 4x16, C/D 16x16 fp32.
// ---------------------------------------------------------------------------
__global__ __launch_bounds__(256)
void gemm_wmma_f32(const float* __restrict__ X, const float* __restrict__ W,
                   const float* __restrict__ bias, float* __restrict__ Y,
                   int nrows) {
    constexpr int LDA = 132;              // row stride % 64 banks == 4 -> no conflicts
    __shared__ float Alds[16 * LDA];

    const int lane    = threadIdx.x & 31;
    const int wave    = threadIdx.x >> 5;
    const int lr      = lane & 15;        // row-within-half / N index
    const int halfsel = (lane < 16) ? 0 : 2;  // K sub-offset per ISA A/B layout
    const int n0      = wave * 16;

    // Preload B fragments: B[k][n] = W[n][k].
    // v0 holds K = 4s+halfsel, v1 holds K = 4s+halfsel+1, at column n0+lr.
    v2f bfrag[32];
    {
        const float* wrow = W + (size_t)(n0 + lr) * IN_CH;
        #pragma unroll
        for (int s = 0; s < 32; ++s) {
            bfrag[s] = *(const v2f*)(wrow + 4 * s + halfsel);
        }
    }

    const int ntiles = nrows >> 4;        // nrows is a multiple of 16 (100000)
    for (int tile = blockIdx.x; tile < ntiles; tile += gridDim.x) {
        const int row0 = tile << 4;

        __syncthreads();
        {   // cooperative 16x128 A-tile load: 2048 floats, 8 per thread
            int e = threadIdx.x * 8;
            int r = e >> 7, c = e & 127;
            const float4* src = (const float4*)(X + (size_t)(row0 + r) * IN_CH + c);
            float4 d0 = src[0];
            float4 d1 = src[1];
            float* dst = Alds + r * LDA + c;
            dst[0] = d0.x; dst[1] = d0.y; dst[2] = d0.z; dst[3] = d0.w;
            dst[4] = d1.x; dst[5] = d1.y; dst[6] = d1.z; dst[7] = d1.w;
        }
        __syncthreads();

        v8f acc = {};
        #pragma unroll
        for (int s = 0; s < 32; ++s) {
            // A frag: lanes 0-15 -> M=lane, K=4s,4s+1; lanes 16-31 -> M=lane-16, K=4s+2,4s+3
            v2f afrag = *(const v2f*)(Alds + lr * LDA + 4 * s + halfsel);
            acc = __builtin_amdgcn_wmma_f32_16x16x4_f32(
                /*neg_a=*/false, afrag, /*neg_b=*/false, bfrag[s],
                /*c_mod=*/(short)0, acc, /*reuse_a=*/false, /*reuse_b=*/false);
        }

        // D layout: VGPR j -> M = j (lanes 0-15) or j+8 (lanes 16-31), N = lane&15
        const int   col   = n0 + lr;
        const float badd  = bias ? bias[col] : 0.0f;
        const int   rbase = row0 + ((lane < 16) ? 0 : 8);
        #pragma unroll
        for (int j = 0; j < 8; ++j) {
            Y[(size_t)(rbase + j) * IN_CH + col] = acc[j] + badd;
        }
    }
}

// ---------------------------------------------------------------------------
// One wave per edge; each lane handles 4 channels (float4 gather + 4 fp32
// hardware atomics into the accumulator). Degree counts every edge; the
// feature scatter is skipped when the gate is zero (~50% of edges).
// ---------------------------------------------------------------------------
__global__ __launch_bounds__(256)
void edge_scatter(const float* __restrict__ xl, const float* __restrict__ alpha,
                  const long long* __restrict__ src_idx,
                  const long long* __restrict__ dst_idx,
                  float* __restrict__ summed, float* __restrict__ deg,
                  int nedges) {
    long long t  = (long long)blockIdx.x * blockDim.x + threadIdx.x;
    int e        = (int)(t >> 5);
    int lane     = (int)(t & 31);
    if (e >= nedges) return;

    const int d = (int)dst_idx[e];
    if (lane == 0) fatomic_add(&deg[d], 1.0f);

    float g = alpha[e] - 0.5f;
    if (g <= 0.0f) return;          // relu gate is zero -> no message
    g *= 2.0f;

    const int s = (int)src_idx[e];
    const float4 v = *(const float4*)(xl + (size_t)s * IN_CH + lane * 4);
    float* o = summed + (size_t)d * IN_CH + lane * 4;
    fatomic_add(o + 0, v.x * g);
    fatomic_add(o + 1, v.y * g);
    fatomic_add(o + 2, v.z * g);
    fatomic_add(o + 3, v.w * g);
}

// out[t] += summed[t] / max(deg[node], 1)   (out already holds bias GEMM)
__global__ __launch_bounds__(256)
void finalize_mean(const float* __restrict__ summed, const float* __restrict__ deg,
                   float* __restrict__ out, int total) {
    int t = blockIdx.x * blockDim.x + threadIdx.x;
    if (t >= total) return;
    float dv = deg[t >> 7];                 // node = t / 128
    out[t] += summed[t] / fmaxf(dv, 1.0f);
}

extern "C" void kernel_launch(void* const* d_in, const int* in_sizes, int n_in,
                              void* d_out, int out_size, void* d_ws, size_t ws_size,
                              hipStream_t stream) {
    (void)in_sizes; (void)n_in; (void)out_size; (void)ws_size;

    const float*     x      = (const float*)d_in[0];
    const float*     alpha  = (const float*)d_in[1];
    const long long* eidx   = (const long long*)d_in[2];   // int64 [2, E]
    const float*     W_lin  = (const float*)d_in[3];
    const float*     W_bias = (const float*)d_in[4];
    const float*     b_bias = (const float*)d_in[5];
    float*           out    = (float*)d_out;

    const long long* src_idx = eidx;
    const long long* dst_idx = eidx + N_EDGES;

    // Workspace layout: x_l [N*128] | summed [N*128] | deg [N]  (~103 MB)
    float* xl     = (float*)d_ws;
    float* summed = xl + (size_t)N_NODES * IN_CH;
    float* deg    = summed + (size_t)N_NODES * IN_CH;

    hipMemsetAsync(summed, 0,
                   ((size_t)N_NODES * IN_CH + N_NODES) * sizeof(float), stream);

    // 1) x_l = x @ W_lin.T   (fp32 WMMA)
    gemm_wmma_f32<<<1024, 256, 0, stream>>>(x, W_lin, nullptr, xl, N_NODES);

    // 2) out = x @ W_bias.T + b_bias  (zero by construction, computed anyway)
    gemm_wmma_f32<<<1024, 256, 0, stream>>>(x, W_bias, b_bias, out, N_NODES);

    // 3) gated scatter-sum + degree
    long long ethreads = (long long)N_EDGES * 32;
    int eblocks = (int)((ethreads + 255) / 256);
    edge_scatter<<<eblocks, 256, 0, stream>>>(xl, alpha, src_idx, dst_idx,
                                              summed, deg, N_EDGES);

    // 4) out += summed / max(deg, 1)
    int total   = N_NODES * IN_CH;
    int fblocks = (total + 255) / 256;
    finalize_mean<<<fblocks, 256, 0, stream>>>(summed, deg, out, total);
}